// AdaptiveGraphRecursiveConvolution_80144089743757
// MI455X (gfx1250) — compile-verified
//
#include <hip/hip_runtime.h>

// Problem constants (match reference)
#define NN 40000
#define GG 2
#define KK 2
#define SS (GG * KK)     // 4 supports*hops
#define EE 640000        // edges per (support,hop)
#define FF 128           // input feature dim
#define OO 128           // output dim
#define KD (2 * FF)      // combined K dim: [h || x] = 256

typedef __attribute__((ext_vector_type(2))) float v2f;
typedef __attribute__((ext_vector_type(8))) float v8f;

// ---------------------------------------------------------------------------
// Zero the f32 output accumulator (harness poisons d_out with 0xAA).
// ---------------------------------------------------------------------------
__global__ void agrc_zero_f32(float* __restrict__ out, int n4) {
    int i = blockIdx.x * blockDim.x + threadIdx.x;
    if (i < n4) {
        float4 z = {0.f, 0.f, 0.f, 0.f};
        ((float4*)out)[i] = z;
    }
}

// ---------------------------------------------------------------------------
// Build combined, pre-scaled, TRANSPOSED weights:
//   WcT[s][o][k]      = mix_w[s/K]     * W[s][k][o]        (k <  FF)
//   WcT[s][o][FF+k]   = inp_mix_w[s/K] * inp_W[s][k][o]    (k <  FF)
// Transposed layout makes the WMMA B-fragment a contiguous b64 load.
// Folding the support-mixing scalars here lets us do ONE SpMM pass total.
// ---------------------------------------------------------------------------
__global__ void agrc_build_wc(const float* __restrict__ W,
                              const float* __restrict__ inpW,
                              const float* __restrict__ mixw,
                              const float* __restrict__ inpmixw,
                              float* __restrict__ WcT) {
    int idx = blockIdx.x * blockDim.x + threadIdx.x;   // over SS*KD*OO
    if (idx >= SS * KD * OO) return;
    int s = idx / (KD * OO);
    int r = idx - s * (KD * OO);
    int f = r / OO;          // k index in [0, KD)
    int o = r - f * OO;
    int g = s / KK;
    float v;
    if (f < FF) {
        v = mixw[g]    * W[((size_t)s * FF + f) * OO + o];
    } else {
        v = inpmixw[g] * inpW[((size_t)s * FF + (f - FF)) * OO + o];
    }
    WcT[((size_t)s * OO + o) * KD + f] = v;
}

// ---------------------------------------------------------------------------
// WMMA GEMM:  pre2[s] = [h || x] @ Wc[s]    ([N,256] @ [256,128] -> [N,128])
// One wave computes a 32x16 output tile (two stacked 16x16 accumulators) so
// each B fragment feeds TWO V_WMMA_F32_16X16X4_F32 ops. All fragment loads
// (A0, A1, B) are contiguous 8-byte loads. 128 WMMAs per wave.
// Block = 256 threads = 8 waves; the 8 waves cover all 8 column tiles (O=128),
// so each block finishes a full 32-row stripe of pre2 for one support s.
//
// Fragment layouts per ISA 7.12.2 (f32, wave32):
//   A 16x4 : lane L -> M = L&15 ; v0/v1 = K {0,1} (L<16) or {2,3} (L>=16)
//   B 4x16 : mirrored (row-striped across lanes)
//   C 16x16: lane L -> N = L&15 ; VGPR r -> M = r (L<16) or 8+r (L>=16)
// ---------------------------------------------------------------------------
__global__ void __launch_bounds__(256)
agrc_gemm_wmma(const float* __restrict__ h,
               const float* __restrict__ x,
               const float* __restrict__ WcT,
               float* __restrict__ pre2) {
    const int blk  = blockIdx.x;          // [0, SS * (NN/32))
    const int s    = blk / (NN / 32);
    const int rt   = blk - s * (NN / 32); // 32-row tile index
    const int wave = threadIdx.x >> 5;    // column tile 0..7
    const int lane = threadIdx.x & 31;

    const int m    = lane & 15;
    const int klo  = (lane & 16) ? 2 : 0;
    const int row0 = rt * 32 + m;         // rows of accumulator 0
    const int row1 = row0 + 16;           // rows of accumulator 1
    const int c0   = wave * 16;

    const float* __restrict__ ha0 = h + (size_t)row0 * FF;
    const float* __restrict__ ha1 = h + (size_t)row1 * FF;
    const float* __restrict__ xa0 = x + (size_t)row0 * FF;
    const float* __restrict__ xa1 = x + (size_t)row1 * FF;
    // B column for this lane: WcT[s][c0+m][*], contiguous in k
    const float* __restrict__ wb0 =
        WcT + ((size_t)s * OO + (c0 + m)) * KD;

    v8f acc0 = {};
    v8f acc1 = {};

    // phase 0: K in [0,128) from h ; phase 1: K in [128,256) from x
    #pragma unroll
    for (int phase = 0; phase < 2; ++phase) {
        const float* __restrict__ i0 = phase ? xa0 : ha0;
        const float* __restrict__ i1 = phase ? xa1 : ha1;
        const float* __restrict__ wb = wb0 + phase * FF;
        #pragma unroll 8
        for (int kk = 0; kk < FF; kk += 4) {
            v2f a0, a1, b;
            a0.x = i0[kk + klo];
            a0.y = i0[kk + klo + 1];
            a1.x = i1[kk + klo];
            a1.y = i1[kk + klo + 1];
            b.x  = wb[kk + klo];
            b.y  = wb[kk + klo + 1];
            acc0 = __builtin_amdgcn_wmma_f32_16x16x4_f32(
                false, a0, false, b, (short)0, acc0, false, false);
            acc1 = __builtin_amdgcn_wmma_f32_16x16x4_f32(
                false, a1, false, b, (short)0, acc1, false, false);
        }
    }

    // Store D tiles: lane<16 -> rows M=0..7, lane>=16 -> rows M=8..15
    const int mr0 = (lane & 16) ? 8 : 0;
    float* op0 = pre2 + (((size_t)s * NN + (size_t)rt * 32) * OO) + c0 + m;
    float* op1 = op0 + (size_t)16 * OO;
    #pragma unroll
    for (int r = 0; r < 8; ++r) {
        op0[(size_t)(mr0 + r) * OO] = acc0[r];
        op1[(size_t)(mr0 + r) * OO] = acc1[r];
    }
}

// ---------------------------------------------------------------------------
// Fused SpMM over all 4 supports into one accumulator:
//   out[dst] += val * pre2[s][src]     (support mixing already folded into Wc)
// One wave per edge; each lane handles 4 features (float4 gather, 4 atomic
// f32 adds). pre2 (82 MB) + out (20 MB) are L2-resident on MI455X (192 MB L2).
// ---------------------------------------------------------------------------
__global__ void __launch_bounds__(256)
agrc_spmm_atomic(const float* __restrict__ pre2,
                 const int* __restrict__ esrc,
                 const int* __restrict__ edst,
                 const float* __restrict__ eval,
                 float* __restrict__ out) {
    int e = blockIdx.x * (blockDim.x >> 5) + (threadIdx.x >> 5);
    if (e >= SS * EE) return;
    const int lane = threadIdx.x & 31;

    const int   srcn = esrc[e];
    const int   dstn = edst[e];
    const float v    = eval[e];
    const int   s    = e / EE;

    const float4* p =
        (const float4*)(pre2 + ((size_t)s * NN + (size_t)srcn) * OO) + lane;
    float4 mfeat = *p;

    float* o = out + (size_t)dstn * OO + lane * 4;
    atomicAdd(o + 0, mfeat.x * v);
    atomicAdd(o + 1, mfeat.y * v);
    atomicAdd(o + 2, mfeat.z * v);
    atomicAdd(o + 3, mfeat.w * v);
}

// ---------------------------------------------------------------------------
// Final ReLU in place on d_out.
// ---------------------------------------------------------------------------
__global__ void agrc_relu(float* __restrict__ out, int n4) {
    int i = blockIdx.x * blockDim.x + threadIdx.x;
    if (i < n4) {
        float4 t = ((float4*)out)[i];
        t.x = fmaxf(t.x, 0.f);
        t.y = fmaxf(t.y, 0.f);
        t.z = fmaxf(t.z, 0.f);
        t.w = fmaxf(t.w, 0.f);
        ((float4*)out)[i] = t;
    }
}

// ---------------------------------------------------------------------------
// Launcher. Inputs (setup_inputs order):
//   0: h [N,F] f32        1: x [N,F] f32
//   2: W [G,K,F,O] f32    3: inp_W [G,K,F,O] f32
//   4: mix_w [G] f32      5: inp_mix_w [G] f32
//   6: edge_val [G,K,E]   7: edge_src [G,K,E] i32   8: edge_dst [G,K,E] i32
// Output: [N,O] f32. Workspace: WcT (512 KB) + pre2 (82 MB).
// ---------------------------------------------------------------------------
extern "C" void kernel_launch(void* const* d_in, const int* in_sizes, int n_in,
                              void* d_out, int out_size, void* d_ws, size_t ws_size,
                              hipStream_t stream) {
    const float* h       = (const float*)d_in[0];
    const float* x       = (const float*)d_in[1];
    const float* W       = (const float*)d_in[2];
    const float* inpW    = (const float*)d_in[3];
    const float* mixw    = (const float*)d_in[4];
    const float* inpmixw = (const float*)d_in[5];
    const float* eval    = (const float*)d_in[6];
    const int*   esrc    = (const int*)d_in[7];
    const int*   edst    = (const int*)d_in[8];
    float*       out     = (float*)d_out;

    float* WcT  = (float*)d_ws;                               // [SS][OO][KD]
    float* pre2 = WcT + (size_t)SS * KD * OO;                 // [SS][NN][OO]

    // 1) zero output accumulator
    {
        int n4 = (NN * OO) / 4;
        agrc_zero_f32<<<(n4 + 255) / 256, 256, 0, stream>>>(out, n4);
    }
    // 2) build scaled, transposed combined weights
    {
        int n = SS * KD * OO;
        agrc_build_wc<<<(n + 255) / 256, 256, 0, stream>>>(W, inpW, mixw, inpmixw, WcT);
    }
    // 3) WMMA GEMM -> pre2
    {
        int blocks = SS * (NN / 32);   // 5000 blocks, 8 waves each
        agrc_gemm_wmma<<<blocks, 256, 0, stream>>>(h, x, WcT, pre2);
    }
    // 4) fused SpMM with f32 atomics
    {
        int edges = SS * EE;
        int epb   = 256 / 32;          // 8 edges per block
        agrc_spmm_atomic<<<(edges + epb - 1) / epb, 256, 0, stream>>>(
            pre2, esrc, edst, eval, out);
    }
    // 5) ReLU
    {
        int n4 = (NN * OO) / 4;
        agrc_relu<<<(n4 + 255) / 256, 256, 0, stream>>>(out, n4);
    }
}